// DCNv2Block_41566693490937
// MI455X (gfx1250) — compile-verified
//
#include <hip/hip_runtime.h>

typedef float v2f __attribute__((ext_vector_type(2)));
typedef float v8f __attribute__((ext_vector_type(8)));
typedef unsigned int v4u __attribute__((ext_vector_type(4)));
typedef int v4i_t __attribute__((ext_vector_type(4)));
typedef int v8i_t __attribute__((ext_vector_type(8)));

#define B_     8
#define C1IN   256
#define C1OUT  128
#define HD1    66                 // DCN1 output spatial dim
#define H1HW   (HD1*HD1)          // 4356
#define NPIX1  (B_*H1HW)          // 34848 (divisible by 32)
#define XHW    4096               // 64*64
#define C2IN   128
#define C2OUT  256
#define KDIM2  (C2IN*9)           // 1152
#define NPIX2  (B_*XHW)           // 32768
#define KP     36                 // padded LDS row stride (32 data + 4 pad dwords)

#if __has_builtin(__builtin_amdgcn_tensor_load_to_lds) && __has_builtin(__builtin_amdgcn_s_wait_tensorcnt)
#define USE_TDM 1
#endif

// ---------- raw LDS byte offset (addrspace(3) value) for TDM descriptors ----------
__device__ __forceinline__ unsigned lds_off(const void* p) {
    return (unsigned)(uintptr_t)(const __attribute__((address_space(3))) void*)p;
}

#ifdef USE_TDM
// 2D fp32 tile load Global->LDS via Tensor Data Mover, with D# padding producing
// KP(36)-dword LDS rows: pad_enable, pad_interval=32 dwords (enc 4), pad_amount=4 dwords (enc 3).
__device__ __forceinline__ void tdm_load_tile_2d(unsigned lds_byte, const float* gsrc,
                                                 unsigned tdim0, unsigned tdim1,
                                                 unsigned tile0, unsigned tile1,
                                                 unsigned stride0) {
    unsigned long long ga = (unsigned long long)(uintptr_t)gsrc;
    v4u g0;
    g0[0] = 1u;                                              // count=1 (valid), user mode
    g0[1] = lds_byte;                                        // lds_addr [63:32]
    g0[2] = (unsigned)(ga & 0xFFFFFFFFu);                    // global_addr lo
    g0[3] = (unsigned)((ga >> 32) & 0x01FFFFFFu) | (2u << 30); // global_addr hi + type=2
    v8i_t g1;
    g1[0] = (int)((2u << 16) | (1u << 20) | (4u << 22) | (3u << 25)); // data_size=4B, pad cfg
    g1[1] = (int)((tdim0 & 0xFFFFu) << 16);                  // tensor_dim0[15:0] @ bits 63:48
    g1[2] = (int)(((tdim0 >> 16) & 0xFFFFu) | ((tdim1 & 0xFFFFu) << 16));
    g1[3] = (int)(((tdim1 >> 16) & 0xFFFFu) | ((tile0 & 0xFFFFu) << 16));
    g1[4] = (int)(tile1 & 0xFFFFu);                          // tile_dim1; tile_dim2=0
    g1[5] = (int)stride0;                                    // tensor_dim0_stride lo32
    g1[6] = 0;
    g1[7] = 0;
    v4i_t z4 = {0, 0, 0, 0};
#if __clang_major__ >= 23
    v8i_t z8 = {0, 0, 0, 0, 0, 0, 0, 0};
    __builtin_amdgcn_tensor_load_to_lds(g0, g1, z4, z4, z8, 0);
#else
    __builtin_amdgcn_tensor_load_to_lds(g0, g1, z4, z4, 0);
#endif
}
#endif

// ---------- bilinear sampling record: 4 clamped indices + 4 weights*mask*valid ----------
__device__ __forceinline__ void make_rec(float py, float px, int H, int W, float mask,
                                         int* idx, float* wt) {
    float y0f = floorf(py), x0f = floorf(px);
    int   y0  = (int)y0f,   x0  = (int)x0f;
    float wy = py - y0f, wx = px - x0f;
    float cw[4] = {(1.f-wy)*(1.f-wx), (1.f-wy)*wx, wy*(1.f-wx), wy*wx};
    const int dy[4] = {0,0,1,1}, dx[4] = {0,1,0,1};
#pragma unroll
    for (int c = 0; c < 4; ++c) {
        int yy = y0 + dy[c], xx = x0 + dx[c];
        bool valid = (yy >= 0) & (yy <= H-1) & (xx >= 0) & (xx <= W-1);
        int yc = yy < 0 ? 0 : (yy > H-1 ? H-1 : yy);
        int xc = xx < 0 ? 0 : (xx > W-1 ? W-1 : xx);
        idx[c] = yc * W + xc;
        wt[c]  = valid ? cw[c] * mask : 0.f;
    }
}

// ---------------- Kernel 1: DCN1 offsets (1x1 conv, pad=1) + sampling records ----------------
__global__ __launch_bounds__(256) void offs1_kernel(
    const float* __restrict__ x, const float* __restrict__ ow1,
    const float* __restrict__ ob1, int4* __restrict__ rIdx, float4* __restrict__ rW) {
    int p = blockIdx.x * 256 + threadIdx.x;
    if (p >= NPIX1) return;
    int b  = p / H1HW, pos = p - b * H1HW;
    int oy = pos / HD1, ox = pos - oy * HD1;
    int iy = oy - 1, ix = ox - 1;             // pad=1, 1x1 kernel
    float s0 = ob1[0], s1 = ob1[1], s2 = ob1[2];
    if (iy >= 0 && iy < 64 && ix >= 0 && ix < 64) {
        const float* xp = x + (size_t)b * C1IN * XHW + iy * 64 + ix;
        for (int c = 0; c < C1IN; ++c) {
            float xv = xp[(size_t)c * XHW];
            s0 = fmaf(xv, ow1[c],          s0);
            s1 = fmaf(xv, ow1[C1IN + c],   s1);
            s2 = fmaf(xv, ow1[2*C1IN + c], s2);
        }
    }
    float mask = 1.f / (1.f + __expf(-s2));
    int idx[4]; float wt[4];
    make_rec(s0 + (float)iy, s1 + (float)ix, 64, 64, mask, idx, wt);
    rIdx[p] = make_int4(idx[0], idx[1], idx[2], idx[3]);
    rW[p]   = make_float4(wt[0], wt[1], wt[2], wt[3]);
}

// ------- shared GEMM helpers: k-contiguous padded rows, b64 fragments, b128 gather commit ----
__device__ __forceinline__ void stage_w_fallback(float* Wl, const float* __restrict__ wsrc,
                                                 int rowStride, int k0, int t) {
#pragma unroll
    for (int i = 0; i < 4; ++i) {              // 4*256 threads * float4 = 4096 floats
        int lin = t + i * 256;
        int m = lin >> 3, kk = (lin & 7) << 2;
        float4 wv = *reinterpret_cast<const float4*>(&wsrc[(size_t)m * rowStride + k0 + kk]);
        *reinterpret_cast<float4*>(&Wl[m * KP + kk]) = wv;
    }
}

// ---------------- Kernel 2: DCN1 GEMM  (M=128, K=256, N=34848) via V_WMMA_F32_16X16X4_F32 ----
__global__ __launch_bounds__(256) void dcn1_gemm_kernel(
    const float* __restrict__ x, const float* __restrict__ w1,
    const float* __restrict__ b1, const int4* __restrict__ rIdx,
    const float4* __restrict__ rW, float* __restrict__ h1) {
    __shared__ float Wl[C1OUT * KP];          // [m][k] padded rows
    __shared__ float Vl[32 * KP];             // [n][k] padded rows
    int t = threadIdx.x, lane = t & 31, wave = t >> 5;
    int p0 = blockIdx.x * 32;
    int p  = p0 + lane;                        // this lane's gather pixel
    int4   ri = rIdx[p];
    float4 rw = rW[p];
    int cb = (p / H1HW) * C1IN;                // b*256 channel base
    v8f acc0 = {}, acc1 = {};
    for (int k0 = 0; k0 < C1IN; k0 += 32) {
        // --- stage W chunk: rows m=0..127, k0..k0+31, into KP-padded LDS rows ---
#ifdef USE_TDM
        if (wave == 0)
            tdm_load_tile_2d(lds_off(Wl), w1 + k0, C1IN, C1OUT, 32, C1OUT, C1IN);
#else
        stage_w_fallback(Wl, w1, C1IN, k0, t);
        if (k0 + 32 < C1IN)
            __builtin_prefetch(&w1[(t >> 1) * C1IN + k0 + 32], 0, 3);
#endif
        // --- gather V chunk (overlaps TDM): row = lane(pixel), k = wave*4..+3 ---
        float vv[4];
#pragma unroll
        for (int q = 0; q < 4; ++q) {
            int kk = (wave << 2) + q;
            const float* src = x + (size_t)(cb + k0 + kk) * XHW;
            vv[q] = rw.x * src[ri.x] + rw.y * src[ri.y]
                  + rw.z * src[ri.z] + rw.w * src[ri.w];
        }
        *reinterpret_cast<float4*>(&Vl[lane * KP + (wave << 2)]) =
            make_float4(vv[0], vv[1], vv[2], vv[3]);
#ifdef USE_TDM
        if (wave == 0) __builtin_amdgcn_s_wait_tensorcnt(0);
#endif
        __syncthreads();
        // --- compute: per wave one 16-row M tile, two 16-col N tiles ---
        int mrow = (wave << 4) + (lane & 15);
        int n0   = lane & 15;
        int kh   = (lane >> 4) << 1;           // 0 for lanes 0-15, 2 for lanes 16-31
#pragma unroll
        for (int ks = 0; ks < 32; ks += 4) {
            int kr = ks + kh;
            float2 af  = *reinterpret_cast<const float2*>(&Wl[mrow * KP + kr]);
            float2 bf0 = *reinterpret_cast<const float2*>(&Vl[n0 * KP + kr]);
            float2 bf1 = *reinterpret_cast<const float2*>(&Vl[(n0 + 16) * KP + kr]);
            v2f a  = {af.x,  af.y};
            v2f b0 = {bf0.x, bf0.y};
            v2f b1v = {bf1.x, bf1.y};
            acc0 = __builtin_amdgcn_wmma_f32_16x16x4_f32(false, a, false, b0,  (short)0, acc0, false, false);
            acc1 = __builtin_amdgcn_wmma_f32_16x16x4_f32(false, a, false, b1v, (short)0, acc1, false, false);
        }
        __syncthreads();
    }
    // C layout (ISA 7.12.2): VGPR r -> M = r + 8*(lane/16); N = lane%16
#pragma unroll
    for (int nt = 0; nt < 2; ++nt) {
        v8f acc = nt ? acc1 : acc0;
        int pn = p0 + nt * 16 + (lane & 15);
        int bo = pn / H1HW, pos = pn - bo * H1HW;
#pragma unroll
        for (int r = 0; r < 8; ++r) {
            int m = (wave << 4) + r + ((lane >> 4) << 3);
            h1[(size_t)(bo * C1OUT + m) * H1HW + pos] = acc[r] + b1[m];
        }
    }
}

// ---------------- Kernel 3: DCN2 offsets (3x3 conv, pad=0) + 9 sampling records --------------
__global__ __launch_bounds__(256) void offs2_kernel(
    const float* __restrict__ h1, const float* __restrict__ ow2,
    const float* __restrict__ ob2, int4* __restrict__ rIdx2, float4* __restrict__ rW2) {
    int p = blockIdx.x * 256 + threadIdx.x;   // 0..32767
    int b = p >> 12, pix = p & 4095;
    int y = pix >> 6, xo = pix & 63;
    float acc[27];
#pragma unroll
    for (int j = 0; j < 27; ++j) acc[j] = ob2[j];
    const float* hb = h1 + (size_t)b * C2IN * H1HW;
    for (int c = 0; c < C2IN; ++c) {
        const float* hc = hb + (size_t)c * H1HW + y * HD1 + xo;
        float hv[9];
#pragma unroll
        for (int tap = 0; tap < 9; ++tap) hv[tap] = hc[(tap / 3) * HD1 + (tap % 3)];
#pragma unroll
        for (int j = 0; j < 27; ++j) {
            const float* wj = ow2 + (size_t)j * KDIM2 + c * 9;   // uniform -> scalar loads
#pragma unroll
            for (int tap = 0; tap < 9; ++tap) acc[j] = fmaf(hv[tap], wj[tap], acc[j]);
        }
    }
#pragma unroll
    for (int k = 0; k < 9; ++k) {
        float mask = 1.f / (1.f + __expf(-acc[18 + k]));
        float py = acc[k]     + (float)(y  + k / 3);
        float px = acc[9 + k] + (float)(xo + k % 3);
        int idx[4]; float wt[4];
        make_rec(py, px, HD1, HD1, mask, idx, wt);
        int ro = p * 9 + k;
        rIdx2[ro] = make_int4(idx[0], idx[1], idx[2], idx[3]);
        rW2[ro]   = make_float4(wt[0], wt[1], wt[2], wt[3]);
    }
}

// ---------------- Kernel 4: DCN2 GEMM (M=256, K=1152, N=32768) + bias + residual -------------
__global__ __launch_bounds__(256) void dcn2_gemm_kernel(
    const float* __restrict__ x, const float* __restrict__ h1,
    const float* __restrict__ w2, const float* __restrict__ b2,
    const int4* __restrict__ rIdx2, const float4* __restrict__ rW2,
    float* __restrict__ out) {
    __shared__ float  Wl[C2IN * KP];          // 128 m-rows per block (blockIdx.y halves M)
    __shared__ float  Vl[32 * KP];
    __shared__ int4   rIl[32 * 9];
    __shared__ float4 rWl[32 * 9];
    int t = threadIdx.x, lane = t & 31, wave = t >> 5;
    int P0 = blockIdx.x * 32;                 // 4096 % 32 == 0 -> whole block same batch
    int mbase = blockIdx.y * 128;
    int b = P0 >> 12, pix0 = P0 & 4095;
    for (int i = t; i < 288; i += 256) {      // stage 32 px * 9 taps of records
        rIl[i] = rIdx2[P0 * 9 + i];
        rWl[i] = rW2[P0 * 9 + i];
    }
    __syncthreads();
    v8f acc0 = {}, acc1 = {};
    for (int k0 = 0; k0 < KDIM2; k0 += 32) {  // 36 chunks
#ifdef USE_TDM
        if (wave == 0)
            tdm_load_tile_2d(lds_off(Wl), w2 + (size_t)mbase * KDIM2 + k0,
                             KDIM2, C2OUT, 32, 128, KDIM2);
#else
        stage_w_fallback(Wl, w2 + (size_t)mbase * KDIM2, KDIM2, k0, t);
        if (k0 + 32 < KDIM2)
            __builtin_prefetch(&w2[(size_t)(mbase + (t >> 1)) * KDIM2 + k0 + 32], 0, 3);
#endif
        // gather V chunk (overlaps TDM)
        float vv[4];
#pragma unroll
        for (int q = 0; q < 4; ++q) {
            int kk  = (wave << 2) + q;
            int kg  = k0 + kk;
            int c   = kg / 9, tap = kg - c * 9;
            int4   ri = rIl[lane * 9 + tap];
            float4 rw = rWl[lane * 9 + tap];
            const float* src = h1 + (size_t)(b * C2IN + c) * H1HW;
            vv[q] = rw.x * src[ri.x] + rw.y * src[ri.y]
                  + rw.z * src[ri.z] + rw.w * src[ri.w];
        }
        *reinterpret_cast<float4*>(&Vl[lane * KP + (wave << 2)]) =
            make_float4(vv[0], vv[1], vv[2], vv[3]);
#ifdef USE_TDM
        if (wave == 0) __builtin_amdgcn_s_wait_tensorcnt(0);
#endif
        __syncthreads();
        int mrow = (wave << 4) + (lane & 15);
        int n0   = lane & 15;
        int kh   = (lane >> 4) << 1;
#pragma unroll
        for (int ks = 0; ks < 32; ks += 4) {
            int kr = ks + kh;
            float2 af  = *reinterpret_cast<const float2*>(&Wl[mrow * KP + kr]);
            float2 bf0 = *reinterpret_cast<const float2*>(&Vl[n0 * KP + kr]);
            float2 bf1 = *reinterpret_cast<const float2*>(&Vl[(n0 + 16) * KP + kr]);
            v2f a  = {af.x,  af.y};
            v2f b0 = {bf0.x, bf0.y};
            v2f b1v = {bf1.x, bf1.y};
            acc0 = __builtin_amdgcn_wmma_f32_16x16x4_f32(false, a, false, b0,  (short)0, acc0, false, false);
            acc1 = __builtin_amdgcn_wmma_f32_16x16x4_f32(false, a, false, b1v, (short)0, acc1, false, false);
        }
        __syncthreads();
    }
#pragma unroll
    for (int nt = 0; nt < 2; ++nt) {
        v8f acc = nt ? acc1 : acc0;
        int pn = pix0 + nt * 16 + (lane & 15);
#pragma unroll
        for (int r = 0; r < 8; ++r) {
            int m = mbase + (wave << 4) + r + ((lane >> 4) << 3);
            size_t o = (size_t)(b * C2OUT + m) * XHW + pn;
            out[o] = x[o] + b2[m] + acc[r];   // residual + bias fused
        }
    }
}

// --------------------------------- launcher --------------------------------------------------
extern "C" void kernel_launch(void* const* d_in, const int* in_sizes, int n_in,
                              void* d_out, int out_size, void* d_ws, size_t ws_size,
                              hipStream_t stream) {
    const float* x   = (const float*)d_in[0];
    const float* w1  = (const float*)d_in[1];
    const float* b1  = (const float*)d_in[2];
    const float* ow1 = (const float*)d_in[3];
    const float* ob1 = (const float*)d_in[4];
    const float* w2  = (const float*)d_in[5];
    const float* b2  = (const float*)d_in[6];
    const float* ow2 = (const float*)d_in[7];
    const float* ob2 = (const float*)d_in[8];
    float* out = (float*)d_out;

    char* ws = (char*)d_ws;
    float* h1 = (float*)ws;
    size_t off = (size_t)B_ * C1OUT * H1HW * sizeof(float);        // 17.8 MB
    int4*   rIdx1 = (int4*)  (ws + off); off += (size_t)NPIX1 * sizeof(int4);
    float4* rW1   = (float4*)(ws + off); off += (size_t)NPIX1 * sizeof(float4);
    int4*   rIdx2 = (int4*)  (ws + off); off += (size_t)NPIX2 * 9 * sizeof(int4);
    float4* rW2   = (float4*)(ws + off);                           // total ~27 MB

    offs1_kernel    <<<(NPIX1 + 255) / 256, 256, 0, stream>>>(x, ow1, ob1, rIdx1, rW1);
    dcn1_gemm_kernel<<<NPIX1 / 32,          256, 0, stream>>>(x, w1, b1, rIdx1, rW1, h1);
    offs2_kernel    <<<NPIX2 / 256,         256, 0, stream>>>(h1, ow2, ob2, rIdx2, rW2);
    dcn2_gemm_kernel<<<dim3(NPIX2 / 32, 2), 256, 0, stream>>>(x, h1, w2, b2, rIdx2, rW2, out);
}